// EmbeddingControlAdapter_35313221107936
// MI455X (gfx1250) — compile-verified
//
#include <hip/hip_runtime.h>
#include <hip/hip_bf16.h>
#include <math.h>

typedef __attribute__((ext_vector_type(2))) float v2f;
typedef __attribute__((ext_vector_type(4))) float f4;
typedef __attribute__((ext_vector_type(8))) float v8f;

// Y[64,N] = act( (A [+ A2]) [64,K] @ W[K,N] + bias[N] )
// One wave (32 lanes) computes one 16x16 tile of Y via V_WMMA_F32_16X16X4_F32.
// block = (32, 4): threadIdx.y = M-tile (M=64 -> 4 tiles), grid.x = N/16.
// Requires K % 16 == 0, N % 16 == 0 (all shapes here satisfy this).
//
// f32 WMMA lane layout (CDNA5 ISA):
//   A (16x4):  lanes 0-15 hold M=lane with K+0/K+1 in VGPR0/1;
//              lanes 16-31 hold M=lane-16 with K+2/K+3.
//   B (4x16):  row-striped across lanes (symmetric to A).
//   C/D (16x16): VGPR r = row r (lanes 0-15) / row r+8 (lanes 16-31), col = lane&15.
template <bool FUSED, bool ACT>
__global__ void eca_gemm_bias_act(const float* __restrict__ A,
                                  const float* __restrict__ A2,
                                  const float* __restrict__ W,
                                  const float* __restrict__ bias,
                                  float* __restrict__ Y,
                                  int K, int N) {
  const int lane  = threadIdx.x;            // 0..31
  const int mtile = threadIdx.y;            // 0..3
  const int ntile = blockIdx.x;
  const int col   = ntile * 16 + (lane & 15);
  const int khalf = (lane >> 4) * 2;        // lanes 0-15 -> K+0/K+1, lanes 16-31 -> K+2/K+3
  const int arow  = mtile * 16 + (lane & 15);

  // Strided pointers; rows are 64B-aligned (K % 16 == 0), so v2f loads are aligned.
  const float* ap  = A + (size_t)arow * K + khalf;
  const float* ap2 = FUSED ? (A2 + (size_t)arow * K + khalf) : nullptr;
  const float* wp  = W + (size_t)khalf * N + col;
  const size_t wstep = 4 * (size_t)N;

  v8f acc = {};

#pragma unroll 4
  for (int kk = 0; kk < K; kk += 4) {
    v2f a = *reinterpret_cast<const v2f*>(ap);
    if (FUSED) {
      v2f a2 = *reinterpret_cast<const v2f*>(ap2);
      a[0] += a2[0];
      a[1] += a2[1];
      ap2 += 4;
    }
    v2f b;
    b[0] = wp[0];
    b[1] = wp[N];
    // D = A x B + C  (8 args: neg_a, A, neg_b, B, c_mod, C, reuse_a, reuse_b)
    acc = __builtin_amdgcn_wmma_f32_16x16x4_f32(false, a, false, b,
                                                (short)0, acc, false, false);
    ap += 4;
    wp += wstep;
  }

  const float bv   = bias[col];
  const int  rbase = mtile * 16 + (lane >> 4) * 8;  // rows r / r+8 split by half-wave
  float* yp = Y + (size_t)rbase * N + col;
#pragma unroll
  for (int r = 0; r < 8; ++r) {
    float x = acc[r] + bv;
    if (ACT) x = x / (1.0f + __expf(-x));   // silu
    yp[(size_t)r * N] = x;
  }
}

// out[b, c, :, :] = V[b, c]; one block per (b,c); 128-bit non-temporal stores.
// HW4 = H*W/4 (all H*W here are multiples of 4).
__global__ void eca_broadcast_hw(const float* __restrict__ V,
                                 float* __restrict__ out, int HW4) {
  const int bc = blockIdx.x;
  const float s = V[bc];
  f4 v4 = {s, s, s, s};
  f4* o = reinterpret_cast<f4*>(out) + (size_t)bc * HW4;
  for (int i = threadIdx.x; i < HW4; i += blockDim.x) {
    __builtin_nontemporal_store(v4, o + i);
  }
}

extern "C" void kernel_launch(void* const* d_in, const int* in_sizes, int n_in,
                              void* d_out, int out_size, void* d_ws, size_t ws_size,
                              hipStream_t stream) {
  (void)in_sizes; (void)n_in; (void)out_size; (void)ws_size;

  static const int CH[4] = {320, 640, 1280, 1280};
  static const int HW[4] = {64 * 64, 32 * 32, 16 * 16, 8 * 8};
  const int BATCH = 64;

  const float* text = (const float*)d_in[0];   // [64, 384]
  const float* pov  = (const float*)d_in[1];   // [64, 512]

  // Workspace: 5 buffers of [64, 1280] fp32 (1.6 MB total)
  const size_t BUF = (size_t)BATCH * 1280;
  float* ws = (float*)d_ws;
  float* ht = ws + 0 * BUF;
  float* t  = ws + 1 * BUF;
  float* hp = ws + 2 * BUF;
  float* p  = ws + 3 * BUF;
  float* v  = ws + 4 * BUF;

  size_t outOff = 0;
  for (int i = 0; i < 4; ++i) {
    const int C = CH[i];
    const float* tw1 = (const float*)d_in[2 + 4 * i + 0];
    const float* tb1 = (const float*)d_in[2 + 4 * i + 1];
    const float* tw2 = (const float*)d_in[2 + 4 * i + 2];
    const float* tb2 = (const float*)d_in[2 + 4 * i + 3];
    const float* pw1 = (const float*)d_in[18 + 4 * i + 0];
    const float* pb1 = (const float*)d_in[18 + 4 * i + 1];
    const float* pw2 = (const float*)d_in[18 + 4 * i + 2];
    const float* pb2 = (const float*)d_in[18 + 4 * i + 3];
    const float* cw  = (const float*)d_in[34 + 2 * i + 0];
    const float* cb  = (const float*)d_in[34 + 2 * i + 1];

    dim3 blk(32, 4);
    dim3 grd(C / 16);

    // h_t = silu(text @ tw1 + tb1); t = h_t @ tw2 + tb2
    eca_gemm_bias_act<false, true ><<<grd, blk, 0, stream>>>(text, nullptr, tw1, tb1, ht, 384, C);
    eca_gemm_bias_act<false, false><<<grd, blk, 0, stream>>>(ht,   nullptr, tw2, tb2, t,    C,  C);
    // h_p = silu(pov @ pw1 + pb1); p = h_p @ pw2 + pb2
    eca_gemm_bias_act<false, true ><<<grd, blk, 0, stream>>>(pov,  nullptr, pw1, pb1, hp, 512, C);
    eca_gemm_bias_act<false, false><<<grd, blk, 0, stream>>>(hp,   nullptr, pw2, pb2, p,    C,  C);
    // v = (t + p) @ cw + cb   (add fused into A-operand load)
    eca_gemm_bias_act<true,  false><<<grd, blk, 0, stream>>>(t,    p,       cw,  cb,  v,    C,  C);

    // broadcast v[b,c] over [H,W] with NT 128-bit stores (608 MB total >> 192 MB L2)
    eca_broadcast_hw<<<dim3(BATCH * C), 256, 0, stream>>>(
        v, (float*)d_out + outOff, HW[i] / 4);

    outOff += (size_t)BATCH * C * HW[i];
  }
}